// get_model_RRFSegNet_47330539602656
// MI455X (gfx1250) — compile-verified
//
#include <hip/hip_runtime.h>
#include <hip/hip_bf16.h>
#include <math.h>

// ---------------------------------------------------------------------------
// RRFSegNet forward for MI455X (gfx1250, wave32, WMMA f32 16x16x4).
// B=4, N=4096, K=20.  All GEMMs run through v_wmma_f32_16x16x4_f32.
// ---------------------------------------------------------------------------

typedef float v2f __attribute__((ext_vector_type(2)));
typedef float v8f __attribute__((ext_vector_type(8)));

#define EPSF 1e-5f

// ---------------------------------------------------------------------------
// kNN: for each point, keep the 21 smallest d2 = sq_i + sq_j - 2*dot (self
// included, dropped afterwards).  Strict '<' insertion reproduces top_k's
// smallest-index tie-break.  Block = 128 threads, candidate tile in LDS.
// ---------------------------------------------------------------------------
__global__ void knn_kernel(const float* __restrict__ x, int* __restrict__ nn_idx,
                           int Nn, int Kn)
{
    const int b = blockIdx.y;
    const int i = blockIdx.x * blockDim.x + threadIdx.x;
    const float* xb = x + (size_t)b * Nn * 3;

    float xi0 = 0.f, xi1 = 0.f, xi2 = 0.f, sqi = 0.f;
    if (i < Nn) {
        xi0 = xb[i * 3 + 0]; xi1 = xb[i * 3 + 1]; xi2 = xb[i * 3 + 2];
        sqi = xi0 * xi0 + xi1 * xi1 + xi2 * xi2;
    }

    const int NB = 21; // K+1
    float bestd[NB];
    int   besti[NB];
    for (int t = 0; t < NB; ++t) { bestd[t] = 3.4e38f; besti[t] = -1; }

    __shared__ float sx[128][4];
    for (int j0 = 0; j0 < Nn; j0 += 128) {
        int j = j0 + threadIdx.x;
        if (j < Nn) {
            float a = xb[j * 3 + 0], bb = xb[j * 3 + 1], c = xb[j * 3 + 2];
            sx[threadIdx.x][0] = a; sx[threadIdx.x][1] = bb;
            sx[threadIdx.x][2] = c; sx[threadIdx.x][3] = a * a + bb * bb + c * c;
        }
        __syncthreads();
        int jn = Nn - j0; if (jn > 128) jn = 128;
        if (i < Nn) {
            for (int t = 0; t < jn; ++t) {
                float dot = xi0 * sx[t][0] + xi1 * sx[t][1] + xi2 * sx[t][2];
                float d2 = sqi + sx[t][3] - 2.0f * dot;
                if (d2 < bestd[NB - 1]) {
                    int pos = NB - 1;
                    while (pos > 0 && d2 < bestd[pos - 1]) {
                        bestd[pos] = bestd[pos - 1];
                        besti[pos] = besti[pos - 1];
                        --pos;
                    }
                    bestd[pos] = d2;
                    besti[pos] = j0 + t;
                }
            }
        }
        __syncthreads();
    }
    if (i < Nn) {
        for (int k = 0; k < Kn; ++k)
            nn_idx[((size_t)b * Kn + k) * Nn + i] = besti[k + 1];
    }
}

// ---------------------------------------------------------------------------
// Generic WMMA GEMM:  Out[Cout x P] = W[Cout x Cin] @ X'[Cin x P] + bias
// in_mode 0: X' = X (optionally BN+ReLU with per-channel (mean,rstd) stats)
// in_mode 1: X' = relation features built from feat[C0 x P2] + nn_idx
//            (p = (b*N + n)*K + k; channels: central | diff_k | diff_{k+1})
// in_mode 2: X' = concat( gmax broadcast over n (C0 rows) , X rows )
//
// Block = 128 threads (4 waves).  Block tile: 64 rows x 32 columns.
// Wave w computes rows [blockIdx.y*64 + 16w, +16) for both 16-col halves.
// K-loop: 64-deep LDS slices (weights + activations both staged in LDS,
// zero-padded), fully unrolled 16 x (2 wmma) per slice.
// ---------------------------------------------------------------------------
__global__ void gemm_wmma_kernel(
    const float* __restrict__ W, const float* __restrict__ bias,
    const float* __restrict__ X, const float* __restrict__ stats, int transform,
    int in_mode, const float* __restrict__ feat, const int* __restrict__ nnidx,
    int C0, int Kn, int Npts,
    float* __restrict__ Out, int Cout, int Cin, int P)
{
    __shared__ float Xs[64][40];   // 64 ci x 32 cols (stride 40: halves hit
                                   // disjoint bank groups for B-frag loads)
    __shared__ float Ws[64][66];   // 64 out-rows x 64 ci (stride 66, 8B align)

    const int p0     = blockIdx.x * 32;
    const int mblock = blockIdx.y * 64;
    const int wave   = threadIdx.x >> 5;
    const int lane   = threadIdx.x & 31;
    const int m0     = mblock + wave * 16;
    const int P2g    = (in_mode == 1) ? (P / Kn) : P;

    v8f acc0 = {0.f, 0.f, 0.f, 0.f, 0.f, 0.f, 0.f, 0.f};
    v8f acc1 = {0.f, 0.f, 0.f, 0.f, 0.f, 0.f, 0.f, 0.f};

    for (int ci0 = 0; ci0 < Cin; ci0 += 64) {
        // ---- stage weight slice (branchless clamp, zero-pad) -------------
        for (int t = threadIdx.x; t < 64 * 64; t += blockDim.x) {
            int r = t >> 6, c = t & 63;
            int row = mblock + r;
            int ci  = ci0 + c;
            bool ok = (row < Cout) && (ci < Cin);
            size_t idx = ok ? ((size_t)row * Cin + ci) : 0;
            float w = W[idx];
            Ws[r][c] = ok ? w : 0.f;
        }
        // ---- stage activation tile (64 x 32) with fused producer transform
        for (int t = threadIdx.x; t < 64 * 32; t += blockDim.x) {
            int ciL = t >> 5, col = t & 31;
            int ci = ci0 + ciL;
            int p  = p0 + col;
            float v = 0.0f;
            if (ci < Cin && p < P) {
                if (in_mode == 0) {
                    v = X[(size_t)ci * P + p];
                    if (transform) {
                        v = (v - stats[2 * ci]) * stats[2 * ci + 1];
                        v = v > 0.f ? v : 0.f;
                    }
                } else if (in_mode == 1) {
                    int k = p % Kn;
                    int q = p / Kn;         // b*N + n
                    int n = q % Npts;
                    int b = q / Npts;
                    if (ci < C0) {
                        v = feat[(size_t)ci * P2g + q];          // central
                    } else {
                        int c = ci - C0;
                        int kk = k;
                        if (c >= C0) { c -= C0; kk = (k + 1) % Kn; } // rolled diff
                        int j = nnidx[((size_t)b * Kn + kk) * Npts + n];
                        v = feat[(size_t)c * P2g + (size_t)b * Npts + j]
                          - feat[(size_t)c * P2g + q];
                    }
                } else { // in_mode == 2
                    if (ci < C0) {
                        int b = p / Npts;
                        v = feat[(size_t)b * C0 + ci];           // gmax broadcast
                    } else {
                        v = X[(size_t)(ci - C0) * P + p];        // act12 rows
                    }
                }
            }
            Xs[ciL][col] = v;
        }
        __syncthreads();

        // ---- WMMA over this 64-deep slice; no branches, fully unrolled ---
        {
            const int ml   = lane & 15;            // A row within wave tile
            const int mrow = (wave * 16) + ml;     // row within Ws
            const int koff = (lane < 16) ? 0 : 2;  // A/B K striping per half
            const int c0   = ml;                   // B col within half 0
#pragma unroll
            for (int kk = 0; kk < 64; kk += 4) {
                v2f a, b0, b1;
                a.x  = Ws[mrow][kk + koff];
                a.y  = Ws[mrow][kk + koff + 1];
                b0.x = Xs[kk + koff][c0];
                b0.y = Xs[kk + koff + 1][c0];
                b1.x = Xs[kk + koff][c0 + 16];
                b1.y = Xs[kk + koff + 1][c0 + 16];
                acc0 = __builtin_amdgcn_wmma_f32_16x16x4_f32(
                           false, a, false, b0, (short)0, acc0, false, false);
                acc1 = __builtin_amdgcn_wmma_f32_16x16x4_f32(
                           false, a, false, b1, (short)0, acc1, false, false);
            }
        }
        __syncthreads();
    }

    // ---- store: D layout lane L, vgpr v -> row v + 8*(L>=16), col L%16
    if (m0 < Cout) {
        int col = lane & 15;
        int rbase = m0 + ((lane < 16) ? 0 : 8);
        int pA = p0 + col;
        int pB = p0 + col + 16;
#pragma unroll
        for (int v = 0; v < 8; ++v) {
            int row = rbase + v;
            if (row < Cout) {
                float bv = bias[row];
                if (pA < P) Out[(size_t)row * P + pA] = acc0[v] + bv;
                if (pB < P) Out[(size_t)row * P + pB] = acc1[v] + bv;
            }
        }
    }
}

// ---------------------------------------------------------------------------
// Per-channel sum / sumsq reduction (grid.y = channel), then mean/rstd.
// ---------------------------------------------------------------------------
__global__ void zero_kernel(float* p, int n)
{
    int i = blockIdx.x * blockDim.x + threadIdx.x;
    if (i < n) p[i] = 0.f;
}

__global__ void stats_kernel(const float* __restrict__ X, float* __restrict__ sums,
                             int P)
{
    const int c = blockIdx.y;
    const float* row = X + (size_t)c * P;
    float s = 0.f, s2 = 0.f;
    for (int p = blockIdx.x * blockDim.x + threadIdx.x; p < P;
         p += gridDim.x * blockDim.x) {
        float v = row[p];
        s += v; s2 += v * v;
    }
    __shared__ float sh0[256], sh1[256];
    sh0[threadIdx.x] = s; sh1[threadIdx.x] = s2;
    __syncthreads();
    for (int st = 128; st > 0; st >>= 1) {
        if ((int)threadIdx.x < st) {
            sh0[threadIdx.x] += sh0[threadIdx.x + st];
            sh1[threadIdx.x] += sh1[threadIdx.x + st];
        }
        __syncthreads();
    }
    if (threadIdx.x == 0) {
        atomicAdd(&sums[2 * c],     sh0[0]);
        atomicAdd(&sums[2 * c + 1], sh1[0]);
    }
}

__global__ void finalize_stats_kernel(const float* __restrict__ sums,
                                      float* __restrict__ stats, int C, float invCnt)
{
    int c = blockIdx.x * blockDim.x + threadIdx.x;
    if (c < C) {
        float m   = sums[2 * c] * invCnt;
        float var = sums[2 * c + 1] * invCnt - m * m;
        stats[2 * c]     = m;
        stats[2 * c + 1] = rsqrtf(var + EPSF);
    }
}

// max over K (raw values; BN+ReLU commute with max since rstd > 0)
__global__ void maxk_kernel(const float* __restrict__ X, float* __restrict__ Y,
                            int C, int P2, int Kn)
{
    int i = blockIdx.x * blockDim.x + threadIdx.x;
    if (i < C * P2) {
        int c = i / P2, q = i % P2;
        const float* p = X + ((size_t)c * P2 + q) * Kn;
        float m = -3.4e38f;
        for (int k = 0; k < Kn; ++k) m = fmaxf(m, p[k]);
        Y[i] = m;
    }
}

__global__ void bnrelu_kernel(const float* __restrict__ X,
                              const float* __restrict__ stats,
                              float* __restrict__ Y, int C, int P)
{
    int i = blockIdx.x * blockDim.x + threadIdx.x;
    if (i < C * P) {
        int c = i / P;
        float v = (X[i] - stats[2 * c]) * stats[2 * c + 1];
        Y[i] = v > 0.f ? v : 0.f;
    }
}

__global__ void transpose_x_kernel(const float* __restrict__ x,
                                   float* __restrict__ pf, int P2)
{
    int i = blockIdx.x * blockDim.x + threadIdx.x;
    if (i < 3 * P2) {
        int c = i / P2, q = i % P2;
        pf[i] = x[(size_t)q * 3 + c];
    }
}

// gmax[b*Cg + c] = max_n G[c*P2 + b*N + n]
__global__ void gmax_kernel(const float* __restrict__ G, float* __restrict__ gmax,
                            int Cg, int Nn, int P2)
{
    const int c = blockIdx.x, b = blockIdx.y;
    const float* row = G + (size_t)c * P2 + (size_t)b * Nn;
    float m = -3.4e38f;
    for (int n = threadIdx.x; n < Nn; n += blockDim.x) m = fmaxf(m, row[n]);
    __shared__ float sh[256];
    sh[threadIdx.x] = m;
    __syncthreads();
    for (int st = 128; st > 0; st >>= 1) {
        if ((int)threadIdx.x < st) sh[threadIdx.x] = fmaxf(sh[threadIdx.x], sh[threadIdx.x + st]);
        __syncthreads();
    }
    if (threadIdx.x == 0) gmax[(size_t)b * Cg + c] = sh[0];
}

// final BN (no relu), reorder (c, b*N+n) -> (b, c, n)
__global__ void final_kernel(const float* __restrict__ raw9,
                             const float* __restrict__ stats,
                             float* __restrict__ out, int Nn, int P2)
{
    int i = blockIdx.x * blockDim.x + threadIdx.x;
    if (i < 2 * P2) {
        int c = i / P2, q = i % P2;
        int b = q / Nn, n = q % Nn;
        out[((size_t)b * 2 + c) * Nn + n] = (raw9[i] - stats[2 * c]) * stats[2 * c + 1];
    }
}

// ---------------------------------------------------------------------------
// Host side
// ---------------------------------------------------------------------------
static void launch_gemm(const float* W, const float* bias, const float* X,
                        const float* stats, int transform, int in_mode,
                        const float* feat, const int* nnidx,
                        int C0, int Kn, int Npts,
                        float* Out, int Cout, int Cin, int P, hipStream_t stream)
{
    dim3 grid((P + 31) / 32, (Cout + 63) / 64);
    gemm_wmma_kernel<<<grid, 128, 0, stream>>>(W, bias, X, stats, transform,
        in_mode, feat, nnidx, C0, Kn, Npts, Out, Cout, Cin, P);
}

static void run_stats(const float* X, int C, int P, float* sums, float* stats,
                      hipStream_t stream)
{
    zero_kernel<<<(2 * C + 255) / 256, 256, 0, stream>>>(sums, 2 * C);
    int nblk = (P + 2047) / 2048; if (nblk > 256) nblk = 256;
    dim3 g(nblk, C);
    stats_kernel<<<g, 256, 0, stream>>>(X, sums, P);
    finalize_stats_kernel<<<(C + 255) / 256, 256, 0, stream>>>(sums, stats, C, 1.0f / (float)P);
}

extern "C" void kernel_launch(void* const* d_in, const int* in_sizes, int n_in,
                              void* d_out, int out_size, void* d_ws, size_t ws_size,
                              hipStream_t stream)
{
    (void)in_sizes; (void)n_in; (void)out_size; (void)ws_size;

    const int Bn = 4, Nn = 4096, Kn = 20;
    const int P2 = Bn * Nn;        // 16384
    const int P  = P2 * Kn;        // 327680

    const float* x     = (const float*)d_in[0];
    const float* n1_w1 = (const float*)d_in[1];  const float* n1_b1 = (const float*)d_in[2];
    const float* n1_w2 = (const float*)d_in[3];  const float* n1_b2 = (const float*)d_in[4];
    const float* n1_w3 = (const float*)d_in[5];  const float* n1_b3 = (const float*)d_in[6];
    const float* n2_w1 = (const float*)d_in[7];  const float* n2_b1 = (const float*)d_in[8];
    const float* n2_w2 = (const float*)d_in[9];  const float* n2_b2 = (const float*)d_in[10];
    const float* n2_w3 = (const float*)d_in[11]; const float* n2_b3 = (const float*)d_in[12];
    const float* g_w   = (const float*)d_in[13]; const float* g_b   = (const float*)d_in[14];
    const float* e_w1  = (const float*)d_in[15]; const float* e_b1  = (const float*)d_in[16];
    const float* e_w2  = (const float*)d_in[17]; const float* e_b2  = (const float*)d_in[18];
    const float* e_w3  = (const float*)d_in[19]; const float* e_b3  = (const float*)d_in[20];
    float* out = (float*)d_out;

    // workspace carve-up (floats)
    float* ws = (float*)d_ws;
    size_t off = 0;
    auto alloc = [&](size_t n) { float* p = ws + off; off += (n + 63) & ~(size_t)63; return p; };
    float* pf     = alloc((size_t)3 * P2);
    int*   nn     = (int*)alloc((size_t)P);              // B*K*N ints
    float* act12  = alloc((size_t)192 * P2);             // out1_act rows 0-63, out2_act rows 64-191
    float* rawmax = alloc((size_t)128 * P2);
    float* raw3   = alloc((size_t)128 * P2);
    float* raw7   = alloc((size_t)256 * P2);
    float* raw8   = alloc((size_t)64 * P2);
    float* raw9   = alloc((size_t)2 * P2);
    float* gmax   = alloc((size_t)Bn * 1024);
    float* sums   = alloc(2048);
    float* stats  = alloc(2048);
    float* rawA   = alloc((size_t)128 * P);              // big ping
    float* rawB   = alloc((size_t)128 * P);              // big pong
    float* graw   = rawA;                                // 1024*P2 fits in rawA

    // ---- kNN + pf
    transpose_x_kernel<<<(3 * P2 + 255) / 256, 256, 0, stream>>>(x, pf, P2);
    knn_kernel<<<dim3(Nn / 128, Bn), 128, 0, stream>>>(x, nn, Nn, Kn);

    // ---- n1 block
    launch_gemm(n1_w1, n1_b1, nullptr, nullptr, 0, 1, pf, nn, 3, Kn, Nn, rawA, 64, 9, P, stream);
    run_stats(rawA, 64, P, sums, stats, stream);
    launch_gemm(n1_w2, n1_b2, rawA, stats, 1, 0, nullptr, nullptr, 0, Kn, Nn, rawB, 64, 64, P, stream);
    run_stats(rawB, 64, P, sums, stats, stream);
    maxk_kernel<<<(64 * P2 + 255) / 256, 256, 0, stream>>>(rawB, rawmax, 64, P2, Kn);
    launch_gemm(n1_w3, n1_b3, rawmax, stats, 1, 0, nullptr, nullptr, 0, Kn, Nn, raw3, 64, 64, P2, stream);
    run_stats(raw3, 64, P2, sums, stats, stream);
    bnrelu_kernel<<<(64 * P2 + 255) / 256, 256, 0, stream>>>(raw3, stats, act12, 64, P2);

    // ---- n2 block (relation features gathered from out1_act on the fly)
    launch_gemm(n2_w1, n2_b1, nullptr, nullptr, 0, 1, act12, nn, 64, Kn, Nn, rawA, 128, 192, P, stream);
    run_stats(rawA, 128, P, sums, stats, stream);
    launch_gemm(n2_w2, n2_b2, rawA, stats, 1, 0, nullptr, nullptr, 0, Kn, Nn, rawB, 128, 128, P, stream);
    run_stats(rawB, 128, P, sums, stats, stream);
    maxk_kernel<<<(128 * P2 + 255) / 256, 256, 0, stream>>>(rawB, rawmax, 128, P2, Kn);
    launch_gemm(n2_w3, n2_b3, rawmax, stats, 1, 0, nullptr, nullptr, 0, Kn, Nn, raw3, 128, 128, P2, stream);
    run_stats(raw3, 128, P2, sums, stats, stream);
    bnrelu_kernel<<<(128 * P2 + 255) / 256, 256, 0, stream>>>(raw3, stats, act12 + (size_t)64 * P2, 128, P2);

    // ---- global feature: g = max_n (g_w @ [out1;out2] + g_b)
    launch_gemm(g_w, g_b, act12, nullptr, 0, 0, nullptr, nullptr, 0, Kn, Nn, graw, 1024, 192, P2, stream);
    gmax_kernel<<<dim3(1024, Bn), 256, 0, stream>>>(graw, gmax, 1024, Nn, P2);

    // ---- embedding head
    launch_gemm(e_w1, e_b1, act12, nullptr, 0, 2, gmax, nullptr, 1024, Kn, Nn, raw7, 256, 1216, P2, stream);
    run_stats(raw7, 256, P2, sums, stats, stream);
    launch_gemm(e_w2, e_b2, raw7, stats, 1, 0, nullptr, nullptr, 0, Kn, Nn, raw8, 64, 256, P2, stream);
    run_stats(raw8, 64, P2, sums, stats, stream);
    launch_gemm(e_w3, e_b3, raw8, stats, 1, 0, nullptr, nullptr, 0, Kn, Nn, raw9, 2, 64, P2, stream);
    run_stats(raw9, 2, P2, sums, stats, stream);
    final_kernel<<<(2 * P2 + 255) / 256, 256, 0, stream>>>(raw9, stats, out, Nn, P2);
}